// GNN_89653147337259
// MI455X (gfx1250) — compile-verified
//
#include <hip/hip_runtime.h>
#include <hip/hip_bf16.h>

// ---------------------------------------------------------------------------
// ChebConv K=4 on gfx1250 (MI455X).
// prop passes are L2-resident gather/scatter (node features = 51MB << 192MB L2),
// GEMMs use full-precision V_WMMA_F32_16X16X4_F32 (fp32 WMMA path).
// ---------------------------------------------------------------------------

typedef float v2f __attribute__((ext_vector_type(2)));
typedef float v8f __attribute__((ext_vector_type(8)));

#define CH 128          // in/out channels
#define NT 8            // 128 / 16 output-column tiles
#define KSTEPS 32       // 128 / 4 K steps per WMMA tile

// ---- zero a float buffer (float4 grid-stride) ------------------------------
__global__ void zero_kernel(float4* __restrict__ p, long n4) {
    long i = (long)blockIdx.x * blockDim.x + threadIdx.x;
    long stride = (long)gridDim.x * blockDim.x;
    float4 z; z.x = z.y = z.z = z.w = 0.0f;
    for (; i < n4; i += stride) p[i] = z;
}

// ---- degree over source index ----------------------------------------------
__global__ void deg_kernel(const int* __restrict__ src, float* __restrict__ deg, int E) {
    int e = blockIdx.x * blockDim.x + threadIdx.x;
    if (e < E) atomicAdd(deg + src[e], 1.0f);
}

// ---- dis = deg > 0 ? rsqrt(deg) : 0  (in place) ----------------------------
__global__ void rsqrt_kernel(float* __restrict__ deg, int n) {
    int i = blockIdx.x * blockDim.x + threadIdx.x;
    if (i < n) {
        float d = deg[i];
        deg[i] = (d > 0.0f) ? rsqrtf(d) : 0.0f;
    }
}

// ---- prop: one wave32 per edge; lane c covers channels c, c+32, c+64, c+96 -
__global__ void prop_kernel(const float* __restrict__ h,
                            const int*   __restrict__ src,
                            const int*   __restrict__ dst,
                            const int*   __restrict__ eattr,  // [E,2]
                            const float* __restrict__ dis,
                            const float* __restrict__ emb1,   // [6,128]
                            const float* __restrict__ emb2,   // [3,128]
                            float*       __restrict__ out,    // pre-zeroed [N,128]
                            int E) {
    int wave = (int)((blockIdx.x * (long)blockDim.x + threadIdx.x) >> 5);
    int lane = threadIdx.x & 31;
    if (wave >= E) return;
    int s  = src[wave];
    int d  = dst[wave];
    int bt = eattr[2 * wave];
    int bd = eattr[2 * wave + 1];
    float w = -dis[s] * dis[d];                // L = I - D^-1/2 A D^-1/2 off-diag
    const float* hs = h    + (size_t)s  * CH;
    const float* e1 = emb1 + (size_t)bt * CH;
    const float* e2 = emb2 + (size_t)bd * CH;
    float*       od = out  + (size_t)d  * CH;
#pragma unroll
    for (int i = 0; i < 4; ++i) {
        int c = lane + 32 * i;
        atomicAdd(od + c, w * (hs[c] + e1[c] + e2[c]));
    }
}

// ---- Chebyshev recurrence: t = 2*t - prev (float4) -------------------------
__global__ void cheb_kernel(float4* __restrict__ t, const float4* __restrict__ prev, long n4) {
    long i = (long)blockIdx.x * blockDim.x + threadIdx.x;
    if (i < n4) {
        float4 a = t[i], b = prev[i];
        a.x = 2.0f * a.x - b.x;
        a.y = 2.0f * a.y - b.y;
        a.z = 2.0f * a.z - b.z;
        a.w = 2.0f * a.w - b.w;
        t[i] = a;
    }
}

// ---- fp32 WMMA GEMM: out(+)= A[M,128] @ W[128,128] (+ bias on init) --------
// One wave per 16x16 output tile. V_WMMA_F32_16X16X4_F32, K unrolled 32x.
// A 16x4 layout: lane l -> row l%16, K pair 2*(l/16)   (contiguous -> b64 load)
// B 4x16 layout: lane l -> col l%16, K pair 2*(l/16)   (two strided b32 loads)
// C/D layout:    VGPR v, lane l -> (M = v + 8*(l/16), N = l%16)
__global__ void gemm_wmma_kernel(const float* __restrict__ A,
                                 const float* __restrict__ W,
                                 const float* __restrict__ bias,
                                 float*       __restrict__ out,
                                 int M, int initMode) {
    int wid  = (int)((blockIdx.x * (long)blockDim.x + threadIdx.x) >> 5);
    int lane = threadIdx.x & 31;
    int tile_m = wid / NT;
    int tile_n = wid % NT;
    if (tile_m * 16 >= M) return;              // M is a multiple of 16 here

    int half = lane >> 4;                      // 0: lanes 0-15, 1: lanes 16-31
    int l16  = lane & 15;
    int mrow = tile_m * 16 + l16;              // A row held by this lane
    int ncol = tile_n * 16 + l16;              // B/C column held by this lane

    v8f c;
    if (initMode) {
        float b = bias[ncol];
#pragma unroll
        for (int v = 0; v < 8; ++v) c[v] = b;
    } else {
#pragma unroll
        for (int v = 0; v < 8; ++v)
            c[v] = out[(size_t)(tile_m * 16 + v + 8 * half) * CH + ncol];
    }

    const float* arow = A + (size_t)mrow * CH + 2 * half;        // 8B aligned
    const float* bcol = W + (size_t)(2 * half) * CH + ncol;

#pragma unroll
    for (int kk = 0; kk < KSTEPS; ++kk) {
        v2f a = *(const v2f*)(arow + 4 * kk);                    // A[m, 4kk+2h .. +1]
        v2f b;
        b[0] = bcol[(size_t)(4 * kk) * CH];                      // W[4kk+2h    , n]
        b[1] = bcol[(size_t)(4 * kk) * CH + CH];                 // W[4kk+2h + 1, n]
        c = __builtin_amdgcn_wmma_f32_16x16x4_f32(
                /*neg_a=*/false, a, /*neg_b=*/false, b,
                /*c_mod=*/(short)0, c, /*reuse_a=*/false, /*reuse_b=*/false);
    }

#pragma unroll
    for (int v = 0; v < 8; ++v)
        out[(size_t)(tile_m * 16 + v + 8 * half) * CH + ncol] = c[v];
}

// ---------------------------------------------------------------------------
extern "C" void kernel_launch(void* const* d_in, const int* in_sizes, int n_in,
                              void* d_out, int out_size, void* d_ws, size_t ws_size,
                              hipStream_t stream) {
    const float* x      = (const float*)d_in[0];
    const int*   eidx   = (const int*)  d_in[1];   // [2,E]
    const int*   eattr  = (const int*)  d_in[2];   // [E,2]
    const float* weight = (const float*)d_in[3];   // [4,128,128]
    const float* bias   = (const float*)d_in[4];   // [128]
    const float* emb1   = (const float*)d_in[5];   // [6,128]
    const float* emb2   = (const float*)d_in[6];   // [3,128]
    float*       out    = (float*)d_out;           // [N,128]

    const int N = in_sizes[0] / CH;
    const int E = in_sizes[1] / 2;
    const int* src = eidx;
    const int* dst = eidx + E;

    // workspace carve-out: dis[N] + three Tx buffers [N,128]
    char* ws = (char*)d_ws;
    size_t off = 0;
    auto carve = [&](size_t bytes) -> float* {
        float* p = (float*)(ws + off);
        off += (bytes + 255) & ~(size_t)255;
        return p;
    };
    float* dis = carve((size_t)N * sizeof(float));
    float* tx1 = carve((size_t)N * CH * sizeof(float));
    float* tx2 = carve((size_t)N * CH * sizeof(float));
    float* tx3 = carve((size_t)N * CH * sizeof(float));
    (void)ws_size;

    const long nch4   = (long)N * CH / 4;
    const int  ZB     = 2048;                     // grid-stride zero blocks
    const int  edgeBl = (E + 7) / 8;              // 8 edges (waves) per 256-thr block
    const int  gemmBl = ((N / 16) * NT) / 4;      // 4 waves per 128-thr block
    const long cb     = (nch4 + 255) / 256;

    // degree -> dis
    zero_kernel<<<256, 256, 0, stream>>>((float4*)dis, (long)N / 4);
    deg_kernel<<<(E + 255) / 256, 256, 0, stream>>>(src, dis, E);
    rsqrt_kernel<<<(N + 255) / 256, 256, 0, stream>>>(dis, N);

    // Tx1 = prop(x)
    zero_kernel<<<ZB, 256, 0, stream>>>((float4*)tx1, nch4);
    prop_kernel<<<edgeBl, 256, 0, stream>>>(x, src, dst, eattr, dis, emb1, emb2, tx1, E);

    // out = x @ W0 + bias ; out += Tx1 @ W1
    gemm_wmma_kernel<<<gemmBl, 128, 0, stream>>>(x,   weight,              bias, out, N, 1);
    gemm_wmma_kernel<<<gemmBl, 128, 0, stream>>>(tx1, weight + 1 * CH * CH, bias, out, N, 0);

    // Tx2 = 2*prop(Tx1) - x ; out += Tx2 @ W2
    zero_kernel<<<ZB, 256, 0, stream>>>((float4*)tx2, nch4);
    prop_kernel<<<edgeBl, 256, 0, stream>>>(tx1, src, dst, eattr, dis, emb1, emb2, tx2, E);
    cheb_kernel<<<(int)cb, 256, 0, stream>>>((float4*)tx2, (const float4*)x, nch4);
    gemm_wmma_kernel<<<gemmBl, 128, 0, stream>>>(tx2, weight + 2 * CH * CH, bias, out, N, 0);

    // Tx3 = 2*prop(Tx2) - Tx1 ; out += Tx3 @ W3
    zero_kernel<<<ZB, 256, 0, stream>>>((float4*)tx3, nch4);
    prop_kernel<<<edgeBl, 256, 0, stream>>>(tx2, src, dst, eattr, dis, emb1, emb2, tx3, E);
    cheb_kernel<<<(int)cb, 256, 0, stream>>>((float4*)tx3, (const float4*)tx1, nch4);
    gemm_wmma_kernel<<<gemmBl, 128, 0, stream>>>(tx3, weight + 3 * CH * CH, bias, out, N, 0);
}